// SimpleGCNEncoder_31851477467888
// MI455X (gfx1250) — compile-verified
//
#include <hip/hip_runtime.h>
#include <hip/hip_bf16.h>

// ---------------------------------------------------------------------------
// SimpleGCNEncoder on gfx1250 (MI455X, wave32, WMMA f32 + TDM async staging)
//
//   deg/dinv -> h = x@W_enc + b_enc (WMMA)
//   3x: hw = h@W (WMMA); agg = scatter-add(norm * hw[src] -> dst);
//       h = relu(agg + b)
//   pool (atomics) -> mol -> p = mol@Wp + bp (WMMA) -> LayerNorm -> out
// ---------------------------------------------------------------------------

typedef __attribute__((ext_vector_type(2))) float v2f;
typedef __attribute__((ext_vector_type(8))) float v8f;

typedef unsigned int tdm_u32x4 __attribute__((ext_vector_type(4)));
typedef int          tdm_i32x4 __attribute__((ext_vector_type(4)));
typedef int          tdm_i32x8 __attribute__((ext_vector_type(8)));

#define HID 256
#define NODE_DIM 64
#define WAVE 32

#define KC 64            // K-chunk staged in LDS per iteration
#define ASTRIDE 68       // padded LDS row stride for A (avoids 64-bank conflicts)

#if __has_builtin(__builtin_amdgcn_tensor_load_to_lds)
#define HAVE_TDM 1
#else
#define HAVE_TDM 0
#endif

// ---------------------------------------------------------------------------
#if HAVE_TDM
// Issue a 2D TDM load: tile_d1 rows x tile_d0 f32 elements, global row stride
// `stride_elems`, into LDS at byte offset lds_addr. Rows beyond tensor_d1
// are zero-filled by the TDM (native OOB handling). pad_bits ORs pad_enable/
// interval/amount into group1 word0 for padded LDS strides.
__device__ static inline void tdm_load_2d(unsigned lds_addr, const void* gaddr,
                                          unsigned tile_d0, unsigned tile_d1,
                                          unsigned tensor_d0, unsigned tensor_d1,
                                          unsigned long long stride_elems,
                                          unsigned pad_bits) {
  unsigned long long ga = (unsigned long long)(uintptr_t)gaddr;
  tdm_u32x4 g0;
  g0[0] = 1u;                                             // count=1 (valid, user)
  g0[1] = lds_addr;                                       // LDS byte address
  g0[2] = (unsigned)(ga & 0xFFFFFFFFu);                   // global_addr[31:0]
  g0[3] = (unsigned)((ga >> 32) & 0x01FFFFFFu) | (2u << 30); // [56:32] | type=2
  tdm_i32x8 g1;
  g1[0] = (int)((2u << 16) | pad_bits);                   // data_size=4B
  g1[1] = (int)((tensor_d0 & 0xFFFFu) << 16);             // tdim0[15:0] @bit48
  g1[2] = (int)(((tensor_d0 >> 16) & 0xFFFFu) |
                ((tensor_d1 & 0xFFFFu) << 16));           // tdim0[31:16], tdim1[15:0]
  g1[3] = (int)(((tensor_d1 >> 16) & 0xFFFFu) |
                ((tile_d0 & 0xFFFFu) << 16));             // tdim1[31:16], tile0
  g1[4] = (int)(tile_d1 & 0xFFFFu);                       // tile1 (tile2=0)
  g1[5] = (int)(unsigned)(stride_elems & 0xFFFFFFFFu);    // dim0_stride[31:0]
  g1[6] = (int)((stride_elems >> 32) & 0xFFFFu);          // dim0_stride[47:32]
  g1[7] = 0;                                              // dim1_stride=0 (2D)
  tdm_i32x4 z4 = {0, 0, 0, 0};
#if defined(__clang_major__) && (__clang_major__ >= 23)
  tdm_i32x8 z8 = {0, 0, 0, 0, 0, 0, 0, 0};
  __builtin_amdgcn_tensor_load_to_lds(g0, g1, z4, z4, z8, 0);
#else
  __builtin_amdgcn_tensor_load_to_lds(g0, g1, z4, z4, 0);
#endif
}
#endif  // HAVE_TDM

// ---------------------------------------------------------------------------
__global__ void fill_zero_kernel(float* __restrict__ p, long long n) {
  long long i = (long long)blockIdx.x * blockDim.x + threadIdx.x;
  long long stride = (long long)gridDim.x * blockDim.x;
  for (; i < n; i += stride) p[i] = 0.0f;
}

__global__ void degree_kernel(const int* __restrict__ dst, float* __restrict__ deg, int E) {
  int e = blockIdx.x * blockDim.x + threadIdx.x;
  if (e < E) atomicAdd(&deg[dst[e]], 1.0f);
}

__global__ void dinv_kernel(float* __restrict__ deg, int N) {
  int i = blockIdx.x * blockDim.x + threadIdx.x;
  if (i < N) deg[i] = rsqrtf(deg[i] + 1.0f);   // +1 self loop
}

// ---------------------------------------------------------------------------
// f32 WMMA GEMM:  C[M,Nc] = A[M,K] @ B[K,Nc] (+bias) (optional relu)
// Block = 256 threads (8 waves). Wave w owns rows [bx*128+16w,+16),
// cols [by*16,+16). Per K-chunk: TDM stages A (128xKC, padded stride) and
// B (KCx16) into LDS; inner loop is pure ds_load + v_wmma_f32_16x16x4_f32.
//
// V_WMMA_F32_16X16X4_F32 operand layouts (wave32):
//   A (16x4):  lane l -> m = l&15 ; VGPR0/1 = K = 2*(l>>4) + {0,1}
//   B (4x16):  lane l -> n = l&15 ; VGPR0/1 = K = 2*(l>>4) + {0,1}
//   C (16x16): lane l -> n = l&15 ; VGPR r  = M = 8*(l>>4) + r
// Requires K % KC == 0 (K = 128 or 256 here).
// ---------------------------------------------------------------------------
__global__ void wmma_gemm_f32_kernel(const float* __restrict__ A,
                                     const float* __restrict__ B,
                                     const float* __restrict__ bias,
                                     float* __restrict__ C,
                                     int M, int K, int Nc,
                                     int do_bias, int do_relu) {
  __shared__ float As[128][ASTRIDE];   // 34,816 B
  __shared__ float Bs[KC][16];         //  4,096 B

  const int tid  = threadIdx.x;
  const int wave = tid >> 5;
  const int lane = tid & 31;
  const int rowBase = blockIdx.x * 128;
  const int row0 = rowBase + wave * 16;
  const int col0 = blockIdx.y * 16;

  v8f acc = {0.f, 0.f, 0.f, 0.f, 0.f, 0.f, 0.f, 0.f};

  const int m  = lane & 15;         // A row-in-tile == B/C col-in-tile
  const int kk = (lane >> 4) << 1;  // 0 or 2

  const int rowsLeft = M - rowBase;  // > 0 by construction

  for (int k0 = 0; k0 < K; k0 += KC) {
#if HAVE_TDM
    if (wave == 0) {
      // A tile: 128 rows x KC f32; pad 4 dwords after every 64 dwords
      // (pad_enable=1<<20, interval code 5 -> 64 dw, amount code 3 -> 4 dw)
      const unsigned padA = (1u << 20) | (5u << 22) | (3u << 25);
      tdm_load_2d((unsigned)(uintptr_t)&As[0][0],
                  A + (size_t)rowBase * K + k0,
                  KC, 128, KC, (unsigned)rowsLeft,
                  (unsigned long long)K, padA);
      // B tile: KC rows x 16 f32, contiguous in LDS
      tdm_load_2d((unsigned)(uintptr_t)&Bs[0][0],
                  B + (size_t)k0 * Nc + col0,
                  16, KC, 16, KC,
                  (unsigned long long)Nc, 0u);
      __builtin_amdgcn_s_wait_tensorcnt(0);
    }
#else
    // Manual staging fallback (vectorized, coalesced float4)
    {
      const int a4 = KC / 4;  // float4 per A row
      for (int i = tid; i < 128 * a4; i += 256) {
        int r  = i / a4;
        int c4 = i % a4;
        int gr = rowBase + r;
        float4 v = make_float4(0.f, 0.f, 0.f, 0.f);
        if (gr < M) v = *(const float4*)(A + (size_t)gr * K + k0 + c4 * 4);
        *(float4*)&As[r][c4 * 4] = v;
      }
      for (int i = tid; i < KC * 4; i += 256) {  // KC rows x 4 float4
        int r  = i >> 2;
        int c4 = i & 3;
        *(float4*)&Bs[r][c4 * 4] =
            *(const float4*)(B + (size_t)(k0 + r) * Nc + col0 + c4 * 4);
      }
    }
#endif
    if (k0 + KC < K)
      __builtin_prefetch(A + (size_t)row0 * K + k0 + KC, 0, 0);  // global_prefetch_b8
    __syncthreads();

    const float* arow = &As[wave * 16 + m][0];
#pragma unroll
    for (int kc = 0; kc < KC; kc += 4) {
      v2f a, b;
      const float2 av = *(const float2*)(arow + kc + kk);
      a.x = av.x;
      a.y = av.y;
      b.x = Bs[kc + kk][m];
      b.y = Bs[kc + kk + 1][m];
      // (neg_a, A, neg_b, B, c_mod, C, reuse_a, reuse_b)
      acc = __builtin_amdgcn_wmma_f32_16x16x4_f32(false, a, false, b,
                                                  (short)0, acc, false, false);
    }
    __syncthreads();
  }

  // store 8 results per lane
  const int half = (lane >> 4) * 8;
  const int col  = col0 + (lane & 15);
  const float bv = do_bias ? bias[col] : 0.0f;
#pragma unroll
  for (int r = 0; r < 8; ++r) {
    int rr = row0 + half + r;
    if (rr < M) {
      float v = acc[r] + bv;
      if (do_relu) v = fmaxf(v, 0.0f);
      C[(size_t)rr * Nc + col] = v;
    }
  }
}

// ---------------------------------------------------------------------------
// agg[dst] += dinv[src]*dinv[dst] * hw[src]  over E real edges + N self loops.
// ---------------------------------------------------------------------------
__global__ void gcn_scatter_kernel(const int* __restrict__ src,
                                   const int* __restrict__ dst,
                                   const float* __restrict__ dinv,
                                   const float* __restrict__ hw,
                                   float* __restrict__ agg,
                                   int E, int N, int H) {
  const int chunks = H >> 2;
  const long long total = (long long)(E + N) * chunks;
  long long idx = (long long)blockIdx.x * blockDim.x + threadIdx.x;
  const long long stride = (long long)gridDim.x * blockDim.x;
  for (; idx < total; idx += stride) {
    const int c = (int)(idx % chunks);
    const long long e = idx / chunks;
    int s, d;
    if (e < E) { s = src[e]; d = dst[e]; }
    else       { s = d = (int)(e - E); }
    const float nrm = dinv[s] * dinv[d];
    const float4 v = *(const float4*)(hw + (size_t)s * H + c * 4);
    float* out = agg + (size_t)d * H + c * 4;
    atomicAdd(out + 0, v.x * nrm);
    atomicAdd(out + 1, v.y * nrm);
    atomicAdd(out + 2, v.z * nrm);
    atomicAdd(out + 3, v.w * nrm);
  }
}

__global__ void bias_relu_kernel(const float* __restrict__ agg,
                                 const float* __restrict__ b,
                                 float* __restrict__ h,
                                 long long total, int H) {
  long long i = (long long)blockIdx.x * blockDim.x + threadIdx.x;
  long long stride = (long long)gridDim.x * blockDim.x;
  for (; i < total; i += stride) {
    int c = (int)(i % H);
    h[i] = fmaxf(agg[i] + b[c], 0.0f);
  }
}

__global__ void pool_accum_kernel(const float* __restrict__ h,
                                  const int* __restrict__ batch,
                                  float* __restrict__ sums,
                                  int N, int H) {
  const int chunks = H >> 2;
  const long long total = (long long)N * chunks;
  long long idx = (long long)blockIdx.x * blockDim.x + threadIdx.x;
  const long long stride = (long long)gridDim.x * blockDim.x;
  for (; idx < total; idx += stride) {
    const int c = (int)(idx % chunks);
    const int n = (int)(idx / chunks);
    const int g = batch[n];
    const float4 v = *(const float4*)(h + (size_t)n * H + c * 4);
    float* out = sums + (size_t)g * H + c * 4;
    atomicAdd(out + 0, v.x);
    atomicAdd(out + 1, v.y);
    atomicAdd(out + 2, v.z);
    atomicAdd(out + 3, v.w);
  }
}

__global__ void pool_cnt_kernel(const int* __restrict__ batch,
                                float* __restrict__ cnt, int N) {
  int i = blockIdx.x * blockDim.x + threadIdx.x;
  if (i < N) atomicAdd(&cnt[batch[i]], 1.0f);
}

__global__ void pool_mean_kernel(const float* __restrict__ sums,
                                 const float* __restrict__ cnt,
                                 float* __restrict__ mol,
                                 int G, int H) {
  long long i = (long long)blockIdx.x * blockDim.x + threadIdx.x;
  long long total = (long long)G * H;
  if (i < total) {
    int g = (int)(i / H);
    mol[i] = sums[i] / fmaxf(cnt[g], 1.0f);
  }
}

// LayerNorm over NODE_DIM=64, one wave32 per graph (2 feats per lane).
__global__ void layernorm64_kernel(const float* __restrict__ p,
                                   const float* __restrict__ gamma,
                                   const float* __restrict__ beta,
                                   float* __restrict__ out, int G) {
  const int wave = threadIdx.x >> 5;
  const int lane = threadIdx.x & 31;
  const int g = blockIdx.x * (blockDim.x >> 5) + wave;
  if (g >= G) return;
  const float v0 = p[(size_t)g * 64 + lane];
  const float v1 = p[(size_t)g * 64 + 32 + lane];
  float s = v0 + v1;
#pragma unroll
  for (int o = 16; o >= 1; o >>= 1) s += __shfl_xor(s, o, WAVE);
  const float mu = s * (1.0f / 64.0f);
  const float d0 = v0 - mu, d1 = v1 - mu;
  float vs = d0 * d0 + d1 * d1;
#pragma unroll
  for (int o = 16; o >= 1; o >>= 1) vs += __shfl_xor(vs, o, WAVE);
  const float rstd = rsqrtf(vs * (1.0f / 64.0f) + 1e-5f);
  out[(size_t)g * 64 + lane]      = d0 * rstd * gamma[lane]      + beta[lane];
  out[(size_t)g * 64 + 32 + lane] = d1 * rstd * gamma[lane + 32] + beta[lane + 32];
}

// ---------------------------------------------------------------------------
static inline size_t align256(size_t x) { return (x + 255) & ~(size_t)255; }

extern "C" void kernel_launch(void* const* d_in, const int* in_sizes, int n_in,
                              void* d_out, int out_size, void* d_ws, size_t ws_size,
                              hipStream_t stream) {
  const float* x      = (const float*)d_in[0];   // [N,128]
  const int*   edge   = (const int*)  d_in[1];   // [2,E] flat
  const int*   batch  = (const int*)  d_in[2];   // [N]
  const float* W_enc  = (const float*)d_in[3];   // [128,256]
  const float* b_enc  = (const float*)d_in[4];   // [256]
  const float* Ws     = (const float*)d_in[5];   // [3,256,256]
  const float* bs     = (const float*)d_in[6];   // [3,256]
  const float* Wp     = (const float*)d_in[7];   // [256,64]
  const float* bp     = (const float*)d_in[8];   // [64]
  const float* gamma  = (const float*)d_in[9];   // [64]
  const float* beta   = (const float*)d_in[10];  // [64]

  const int E = in_sizes[1] / 2;
  const int N = in_sizes[2];
  const int ATOM = in_sizes[0] / N;              // 128
  const int G = out_size / NODE_DIM;             // 2000
  const int* srcp = edge;
  const int* dstp = edge + E;

  // ---- workspace layout ----
  char* ws = (char*)d_ws;
  size_t off = 0;
  float* dinv = (float*)(ws + off); off = align256(off + (size_t)N * 4);
  float* h    = (float*)(ws + off); off = align256(off + (size_t)N * HID * 4);
  float* hw   = (float*)(ws + off); off = align256(off + (size_t)N * HID * 4);
  float* agg  = (float*)(ws + off); off = align256(off + (size_t)N * HID * 4);
  float* sums = (float*)(ws + off); off = align256(off + (size_t)G * HID * 4);
  float* cnt  = (float*)(ws + off); off = align256(off + (size_t)G * 4);
  float* mol  = (float*)(ws + off); off = align256(off + (size_t)G * HID * 4);
  float* pbuf = (float*)(ws + off); off = align256(off + (size_t)G * NODE_DIM * 4);

  const int TB = 256;
  const int gsBlocks = 2048;

  // ---- degree / dinv ----
  fill_zero_kernel<<<512, TB, 0, stream>>>(dinv, (long long)N);
  degree_kernel<<<(E + TB - 1) / TB, TB, 0, stream>>>(dstp, dinv, E);
  dinv_kernel<<<(N + TB - 1) / TB, TB, 0, stream>>>(dinv, N);

  // ---- encoder: h = x @ W_enc + b_enc ----
  {
    dim3 grid((N + 127) / 128, HID / 16);
    wmma_gemm_f32_kernel<<<grid, TB, 0, stream>>>(x, W_enc, b_enc, h,
                                                  N, ATOM, HID, 1, 0);
  }

  // ---- 3 GCN layers ----
  for (int layer = 0; layer < 3; ++layer) {
    dim3 grid((N + 127) / 128, HID / 16);
    wmma_gemm_f32_kernel<<<grid, TB, 0, stream>>>(h, Ws + (size_t)layer * HID * HID,
                                                  nullptr, hw, N, HID, HID, 0, 0);
    fill_zero_kernel<<<gsBlocks, TB, 0, stream>>>(agg, (long long)N * HID);
    gcn_scatter_kernel<<<gsBlocks, TB, 0, stream>>>(srcp, dstp, dinv, hw, agg, E, N, HID);
    bias_relu_kernel<<<gsBlocks, TB, 0, stream>>>(agg, bs + (size_t)layer * HID, h,
                                                  (long long)N * HID, HID);
  }

  // ---- global mean pool ----
  fill_zero_kernel<<<512, TB, 0, stream>>>(sums, (long long)G * HID);
  fill_zero_kernel<<<64, TB, 0, stream>>>(cnt, (long long)G);
  pool_accum_kernel<<<gsBlocks, TB, 0, stream>>>(h, batch, sums, N, HID);
  pool_cnt_kernel<<<(N + TB - 1) / TB, TB, 0, stream>>>(batch, cnt, N);
  pool_mean_kernel<<<((long long)G * HID + TB - 1) / TB, TB, 0, stream>>>(sums, cnt, mol, G, HID);

  // ---- projection: p = mol @ Wp + bp ----
  {
    dim3 grid((G + 127) / 128, NODE_DIM / 16);
    wmma_gemm_f32_kernel<<<grid, TB, 0, stream>>>(mol, Wp, bp, pbuf,
                                                  G, HID, NODE_DIM, 1, 0);
  }

  // ---- LayerNorm -> d_out ----
  layernorm64_kernel<<<(G + 7) / 8, TB, 0, stream>>>(pbuf, gamma, beta, (float*)d_out, G);
}